// CASSI_73246372266466
// MI455X (gfx1250) — compile-verified
//
#include <hip/hip_runtime.h>
#include <hip/hip_bf16.h>
#include <stdint.h>

// CASSI shear: y2[m, n+l] += x[m,n,l] * ca[m,n]
// x: (1, 1024, 1024, 31) f32, ca: (1, 1024, 1024, 1) f32, out: (1, 1024, 1054, 1) f32
// HBM-bound (AI ~0.47 flop/byte, ~138 MB traffic -> ~6 us floor at 23.3 TB/s).
// CDNA5 strategy: double-buffered async global->LDS staging (ASYNCcnt pipeline),
// conflict-free LDS gather (row stride 31 words, coprime with 64 banks),
// register accumulation across n-tiles, single coalesced store per output element.

#define M_DIM  1024
#define N_DIM  1024
#define L_DIM  31
#define OUTW   (N_DIM + L_DIM - 1)   // 1054
#define TN     256                    // n-tile size
#define NTILES (N_DIM / TN)           // 4
#define BLOCK  512                    // 16 wave32s
#define XCHUNKS (TN * L_DIM / 4)      // 1984 x 16B chunks per x tile
#define XTAIL   (XCHUNKS - 3 * BLOCK) // 448

// CDNA5 async copy: 16 bytes global -> LDS per lane, tracked by ASYNCcnt.
__device__ __forceinline__ void async_copy_b128(uint32_t lds_off, const void* gaddr) {
    asm volatile("global_load_async_to_lds_b128 %0, %1, off"
                 :: "v"(lds_off), "v"(gaddr)
                 : "memory");
}

__device__ __forceinline__ void wait_async_zero() {
    asm volatile("s_wait_asynccnt 0" ::: "memory");
}

// Issue the async DMA for one n-tile into LDS buffer `buf` (no waiting here).
__device__ __forceinline__ void stage_tile(int buf, int n0, int tid,
                                           const float* xrow, const float* crow,
                                           uint32_t xs_base, uint32_t cs_base) {
    const char*    gx    = (const char*)(xrow + (size_t)n0 * L_DIM);
    const uint32_t lbase = xs_base + (uint32_t)buf * (TN * L_DIM * 4);
    // 1984 chunks = 3 full rounds of 512 lanes + 448 tail: straight-line issues.
#pragma unroll
    for (int r = 0; r < 3; ++r) {
        const int c = tid + r * BLOCK;
        async_copy_b128(lbase + (uint32_t)c * 16u, gx + (size_t)c * 16u);
    }
    if (tid < XTAIL) {
        const int c = tid + 3 * BLOCK;
        async_copy_b128(lbase + (uint32_t)c * 16u, gx + (size_t)c * 16u);
    }
    // ca tile: 256 floats = 64 x 16B chunks.
    if (tid < TN / 4) {
        async_copy_b128(cs_base + (uint32_t)buf * (TN * 4) + (uint32_t)tid * 16u,
                        (const char*)(crow + n0) + (size_t)tid * 16u);
    }
}

// Gather contribution of one n-tile into output column k.
// xs: packed tile x[m, n0:n0+TN, 0:L] (row stride L_DIM floats), cs: ca[m, n0:n0+TN].
__device__ __forceinline__ float tile_gather(const float* xs, const float* cs,
                                             int k, int n0) {
    const int j = k - n0;                 // column relative to tile start
    if (j < 0 || j > TN + L_DIM - 2) return 0.0f;
    const int l_lo = (j - (TN - 1)) > 0 ? (j - (TN - 1)) : 0;
    const int l_hi = (j < L_DIM - 1) ? j : (L_DIM - 1);
    float a = 0.0f;
    for (int l = l_lo; l <= l_hi; ++l) {
        const int n = j - l;              // n in [0, TN)
        a += xs[n * L_DIM + l] * cs[n];   // lane-adjacent k -> addr stride 31 words: no bank conflicts
    }
    return a;
}

__global__ __launch_bounds__(BLOCK)
void CASSI_shear_kernel(const float* __restrict__ x,
                        const float* __restrict__ ca,
                        float* __restrict__ out) {
    __shared__ __align__(16) float xs[2][TN * L_DIM]; // 2 x 31744 B
    __shared__ __align__(16) float cs[2][TN];         // 2 x  1024 B  (total 64 KB)

    const int m   = blockIdx.x;
    const int tid = threadIdx.x;

    const float* xrow = x  + (size_t)m * N_DIM * L_DIM;
    const float* crow = ca + (size_t)m * N_DIM;

    // Low 32 bits of a flat pointer into LDS == LDS byte offset (CDNA5 aperture rule).
    const uint32_t xs_base = (uint32_t)(uintptr_t)(&xs[0][0]);
    const uint32_t cs_base = (uint32_t)(uintptr_t)(&cs[0][0]);

    // Each thread owns output columns tid, tid+512, (tid+1024 if < OUTW).
    float acc0 = 0.0f, acc1 = 0.0f, acc2 = 0.0f;

    // Prime the pipeline: tile 0 -> buffer 0.
    stage_tile(0, 0, tid, xrow, crow, xs_base, cs_base);

    for (int t = 0; t < NTILES; ++t) {
        const int buf = t & 1;
        const int n0  = t * TN;

        wait_async_zero();   // tile t's DMA (issued one iteration ago) has landed
        __syncthreads();     // all waves: data visible AND compute(t-1) finished

        // Prefetch tile t+1 into the other buffer; flows under compute(t).
        if (t + 1 < NTILES)
            stage_tile(buf ^ 1, n0 + TN, tid, xrow, crow, xs_base, cs_base);

        acc0 += tile_gather(xs[buf], cs[buf], tid,             n0);
        acc1 += tile_gather(xs[buf], cs[buf], tid + BLOCK,     n0); // tid+512 < 1054 always
        if (tid < OUTW - 2 * BLOCK)                                  // tid < 30
            acc2 += tile_gather(xs[buf], cs[buf], tid + 2 * BLOCK, n0);
    }

    float* orow = out + (size_t)m * OUTW;
    orow[tid]         = acc0;
    orow[tid + BLOCK] = acc1;
    if (tid < OUTW - 2 * BLOCK)
        orow[tid + 2 * BLOCK] = acc2;
}

extern "C" void kernel_launch(void* const* d_in, const int* in_sizes, int n_in,
                              void* d_out, int out_size, void* d_ws, size_t ws_size,
                              hipStream_t stream) {
    (void)in_sizes; (void)n_in; (void)out_size; (void)d_ws; (void)ws_size;
    const float* x   = (const float*)d_in[0];
    const float* ca  = (const float*)d_in[1];
    float*       out = (float*)d_out;

    dim3 grid(M_DIM);
    dim3 block(BLOCK);
    hipLaunchKernelGGL(CASSI_shear_kernel, grid, block, 0, stream, x, ca, out);
}